// GCNSepModule_10359461118094
// MI455X (gfx1250) — compile-verified
//
#include <hip/hip_runtime.h>
#include <hip/hip_bf16.h>
#include <math.h>

#define N_NODES 50000
#define N_EDGES 600000
#define DIM     128
#define K2      256   // 2*DIM
#define LN_EPS  1e-5f
#define LDS_STRIDE 260  // 260 % 64 == 4 -> conflict-free across 16 lanes

typedef __attribute__((ext_vector_type(2))) float v2f;
typedef __attribute__((ext_vector_type(8))) float v8f;

// ---------------------------------------------------------------------------
// Zero scratch (agg + degree arrays) — must run every launch (graph replay).
// ---------------------------------------------------------------------------
__global__ void zero_kernel(float* __restrict__ p, int n) {
    int i = blockIdx.x * blockDim.x + threadIdx.x;
    int stride = gridDim.x * blockDim.x;
    for (; i < n; i += stride) p[i] = 0.0f;
}

// ---------------------------------------------------------------------------
// Degree counts via f32 atomics (edges are int64 in the reference).
// ---------------------------------------------------------------------------
__global__ void degree_kernel(const long long* __restrict__ ei,
                              float* __restrict__ deg_out,
                              float* __restrict__ deg_in) {
    int e = blockIdx.x * blockDim.x + threadIdx.x;
    if (e >= N_EDGES) return;
    int s = (int)ei[e];
    int d = (int)ei[N_EDGES + e];
    atomicAdd(&deg_out[s], 1.0f);
    atomicAdd(&deg_in[d], 1.0f);
}

// deg -> rsqrt(max(deg,1)) in place (covers deg_out and deg_in contiguously)
__global__ void rsqrt_kernel(float* __restrict__ deg, int n) {
    int i = blockIdx.x * blockDim.x + threadIdx.x;
    if (i < n) {
        float v = deg[i];
        deg[i] = rsqrtf(fmaxf(v, 1.0f));
    }
}

// ---------------------------------------------------------------------------
// LayerNorm: one wave32 per row; each lane owns 4 contiguous floats.
// ---------------------------------------------------------------------------
__global__ __launch_bounds__(256) void layernorm_kernel(
        const float* __restrict__ h, const float* __restrict__ gamma,
        const float* __restrict__ beta, float* __restrict__ hn) {
    int wave = threadIdx.x >> 5;
    int lane = threadIdx.x & 31;
    int row  = blockIdx.x * 8 + wave;
    if (row >= N_NODES) return;

    float4 x = ((const float4*)(h + (size_t)row * DIM))[lane];

    float sum = x.x + x.y + x.z + x.w;
#pragma unroll
    for (int m = 16; m >= 1; m >>= 1) sum += __shfl_xor(sum, m, 32);
    float mu = sum * (1.0f / DIM);

    float cx = x.x - mu, cy = x.y - mu, cz = x.z - mu, cw = x.w - mu;
    float vs = cx * cx + cy * cy + cz * cz + cw * cw;
#pragma unroll
    for (int m = 16; m >= 1; m >>= 1) vs += __shfl_xor(vs, m, 32);
    float rs = rsqrtf(vs * (1.0f / DIM) + LN_EPS);

    float4 g = ((const float4*)gamma)[lane];
    float4 b = ((const float4*)beta)[lane];
    float4 o;
    o.x = cx * rs * g.x + b.x;
    o.y = cy * rs * g.y + b.y;
    o.z = cz * rs * g.z + b.z;
    o.w = cw * rs * g.w + b.w;
    ((float4*)(hn + (size_t)row * DIM))[lane] = o;
}

// ---------------------------------------------------------------------------
// Scatter: one wave32 per edge (uniform src/dst -> scalar loads),
// each lane gathers float4 from hn[src], scales by inv_sqrt_out[src],
// 4x global_atomic_add_f32 into agg[dst]. hn+agg are L2-resident (51MB<192MB).
// ---------------------------------------------------------------------------
__global__ __launch_bounds__(256) void scatter_kernel(
        const long long* __restrict__ ei, const float* __restrict__ hn,
        const float* __restrict__ inv_out, float* __restrict__ agg) {
    int tid  = blockIdx.x * blockDim.x + threadIdx.x;
    int e    = tid >> 5;
    int lane = tid & 31;
    if (e >= N_EDGES) return;
    int s = (int)ei[e];
    int d = (int)ei[N_EDGES + e];
    float sc = inv_out[s];
    float4 v = ((const float4*)(hn + (size_t)s * DIM))[lane];
    float* ap = agg + (size_t)d * DIM + lane * 4;
    atomicAdd(ap + 0, v.x * sc);
    atomicAdd(ap + 1, v.y * sc);
    atomicAdd(ap + 2, v.z * sc);
    atomicAdd(ap + 3, v.w * sc);
}

// ---------------------------------------------------------------------------
// GEMM: out[N,128] = cat([hn, agg*inv_in]) @ W^T + b  using fp32 WMMA.
// Block = 256 threads = 8 waves, one 16-row x 128-col output tile per block.
// A-tile (16 x 256, with inv_sqrt_in scaling fused) staged in LDS once,
// padded stride 260 floats for conflict-free ds_load_b64 fragment reads.
// Per ISA 7.12.2 (f32, 16x16x4):
//   A: lane l(0-15) v0/v1 = A[l][2*kh+0/1]   (kh = lane>>4)
//   B: lane n(0-15) v0/v1 = B[2*kh+0/1][n] = W[col0+n][k0+2*kh+0/1]
//   D: VGPR r -> row r + 8*kh, lane&15 -> column
// ---------------------------------------------------------------------------
__global__ __launch_bounds__(256) void gemm_kernel(
        const float* __restrict__ hn, const float* __restrict__ agg,
        const float* __restrict__ inv_in, const float* __restrict__ W,
        const float* __restrict__ bias, float* __restrict__ out) {
    __shared__ float lds[16 * LDS_STRIDE];

    int row0 = blockIdx.x * 16;
    int t = threadIdx.x;

    // Stage cat-tile: 16 rows x 256 K values, coalesced (thread = K column).
#pragma unroll
    for (int j = 0; j < 16; ++j) {
        int r = row0 + j;
        float v;
        if (t < DIM) v = hn[(size_t)r * DIM + t];
        else         v = agg[(size_t)r * DIM + (t - DIM)] * inv_in[r];
        lds[j * LDS_STRIDE + t] = v;
    }
    __syncthreads();

    int wave = t >> 5;
    int lane = t & 31;
    int lrow = lane & 15;
    int kh   = lane >> 4;
    int col0 = wave * 16;

    const float* ap = lds + lrow * LDS_STRIDE + 2 * kh;
    const float* wp = W + (size_t)(col0 + lrow) * K2 + 2 * kh;

    v8f c = {};
#pragma unroll 8
    for (int k0 = 0; k0 < K2; k0 += 4) {
        v2f a = *(const v2f*)(ap + k0);
        v2f b = *(const v2f*)(wp + k0);
        // V_WMMA_F32_16X16X4_F32: (neg_a, A, neg_b, B, c_mod, C, reuse_a, reuse_b)
        c = __builtin_amdgcn_wmma_f32_16x16x4_f32(
                false, a, false, b, (short)0, c, false, false);
    }

    float bv = bias[col0 + lrow];
#pragma unroll
    for (int r = 0; r < 8; ++r) {
        out[(size_t)(row0 + kh * 8 + r) * DIM + col0 + lrow] = c[r] + bv;
    }
}

// ---------------------------------------------------------------------------
extern "C" void kernel_launch(void* const* d_in, const int* in_sizes, int n_in,
                              void* d_out, int out_size, void* d_ws, size_t ws_size,
                              hipStream_t stream) {
    const float*     h     = (const float*)d_in[0];
    const long long* ei    = (const long long*)d_in[1]; // int64 per reference
    const float*     gamma = (const float*)d_in[2];
    const float*     beta  = (const float*)d_in[3];
    const float*     W     = (const float*)d_in[4];
    const float*     bias  = (const float*)d_in[5];
    float*           out   = (float*)d_out;

    float* ws      = (float*)d_ws;
    float* hn      = ws;                                  // 6,400,000 f32
    float* agg     = hn  + (size_t)N_NODES * DIM;         // 6,400,000 f32
    float* deg_out = agg + (size_t)N_NODES * DIM;         //    50,000 f32 -> inv_sqrt_out
    float* deg_in  = deg_out + N_NODES;                   //    50,000 f32 -> inv_sqrt_in

    // 1) zero agg + both degree arrays (contiguous)
    int nz = N_NODES * DIM + 2 * N_NODES;
    zero_kernel<<<2048, 256, 0, stream>>>(agg, nz);

    // 2) degrees
    degree_kernel<<<(N_EDGES + 255) / 256, 256, 0, stream>>>(ei, deg_out, deg_in);

    // 3) deg -> rsqrt(max(deg,1)), in place for both arrays
    rsqrt_kernel<<<(2 * N_NODES + 255) / 256, 256, 0, stream>>>(deg_out, 2 * N_NODES);

    // 4) layernorm
    layernorm_kernel<<<N_NODES / 8, 256, 0, stream>>>(h, gamma, beta, hn);

    // 5) scatter-add (one wave per edge)
    scatter_kernel<<<(N_EDGES * 32) / 256, 256, 0, stream>>>(ei, hn, deg_out, agg);

    // 6) fp32 WMMA GEMM + bias
    gemm_kernel<<<N_NODES / 16, 256, 0, stream>>>(hn, agg, deg_in, W, bias, out);
}